// GCN_8057358648366
// MI455X (gfx1250) — compile-verified
//
#include <hip/hip_runtime.h>
#include <hip/hip_bf16.h>
#include <math.h>

// ---------------------------------------------------------------------------
// 2-layer GCN on MI455X (gfx1250, wave32).
//   z1 = A @ X            (COO SpMM, atomicAdd f32)
//   h  = relu(z1 @ W1)    (V_WMMA_F32_16X16X4_F32, fp32-exact)
//   z2 = A @ h            (COO SpMM)
//   out= log_softmax(z2 @ W2)   (WMMA + LDS row softmax)
// ---------------------------------------------------------------------------

typedef __attribute__((ext_vector_type(2))) float v2f;
typedef __attribute__((ext_vector_type(8))) float v8f;

#define IN_DIM   128
#define HIDDEN    16
#define NCLASS    40
#define NPAD      48   // NCLASS padded to 3 tiles of 16

// ---------------------------------------------------------------------------
__global__ void gcn_zero_f32(float* __restrict__ p, int n) {
  int i = blockIdx.x * blockDim.x + threadIdx.x;
  if (i < n) p[i] = 0.0f;
}

// ---------------------------------------------------------------------------
// SpMM layer 1: z1[r, :] += val * x[c, :], 128 features.
// 32 threads per edge, each thread owns one float4 chunk (32*4 = 128).
__global__ void gcn_spmm1(const int* __restrict__ rows,
                          const int* __restrict__ cols,
                          const float* __restrict__ vals,
                          const float* __restrict__ x,
                          float* __restrict__ z1, int nEdges) {
  int gid = blockIdx.x * blockDim.x + threadIdx.x;
  int e  = gid >> 5;
  int f4 = gid & 31;
  if (e >= nEdges) return;
  int r = rows[e];
  int c = cols[e];
  float v = vals[e];
  const float4* xr = (const float4*)(x + (size_t)c * IN_DIM);
  float4 d = xr[f4];
  float* zr = z1 + (size_t)r * IN_DIM + (size_t)f4 * 4;
  atomicAdd(zr + 0, v * d.x);
  atomicAdd(zr + 1, v * d.y);
  atomicAdd(zr + 2, v * d.z);
  atomicAdd(zr + 3, v * d.w);
}

// ---------------------------------------------------------------------------
// SpMM layer 2: z2[r, :] += val * h[c, :], 16 features. 4 threads per edge.
__global__ void gcn_spmm2(const int* __restrict__ rows,
                          const int* __restrict__ cols,
                          const float* __restrict__ vals,
                          const float* __restrict__ h,
                          float* __restrict__ z2, int nEdges) {
  int gid = blockIdx.x * blockDim.x + threadIdx.x;
  int e  = gid >> 2;
  int f4 = gid & 3;
  if (e >= nEdges) return;
  int r = rows[e];
  int c = cols[e];
  float v = vals[e];
  const float4* hr = (const float4*)(h + (size_t)c * HIDDEN);
  float4 d = hr[f4];
  float* zr = z2 + (size_t)r * HIDDEN + (size_t)f4 * 4;
  atomicAdd(zr + 0, v * d.x);
  atomicAdd(zr + 1, v * d.y);
  atomicAdd(zr + 2, v * d.z);
  atomicAdd(zr + 3, v * d.w);
}

// ---------------------------------------------------------------------------
// GEMM1 + ReLU: h[tile*16 .. +16, 0..16) = relu(z1_tile[16x128] @ W1[128x16])
// One wave32 per M-tile. fp32 WMMA: D = A(16x4) * B(4x16) + C, 32 K-steps.
//
// f32 A layout (ISA 7.12.2): lanes 0-15 hold row M=lane, VGPR0=K+0, VGPR1=K+1;
// lanes 16-31 hold row M=lane-16, VGPR0=K+2, VGPR1=K+3. B mirrors it with
// N across lanes (row of B striped across lanes within one VGPR).
__global__ void __launch_bounds__(32)
gcn_gemm1_relu(const float* __restrict__ z1,
               const float* __restrict__ w1,
               float* __restrict__ h) {
  __shared__ float Wl[IN_DIM * HIDDEN];  // 128x16, row-major (K major)
  __shared__ float Al[16 * IN_DIM];      // 16x128, row-major (M major)
  const int lane = threadIdx.x;
  const int tile = blockIdx.x;

  for (int i = lane; i < IN_DIM * HIDDEN; i += 32) Wl[i] = w1[i];
  const float* Ab = z1 + (size_t)tile * 16 * IN_DIM;
  for (int i = lane; i < 16 * IN_DIM; i += 32) Al[i] = Ab[i];
  __syncthreads();

  const int half = lane >> 4;    // 0: K+0/K+1 ; 1: K+2/K+3
  const int l    = lane & 15;    // A: row M ; B: col N

  v8f c = {};
  #pragma unroll 8
  for (int k0 = 0; k0 < IN_DIM; k0 += 4) {
    const int ka = k0 + 2 * half;
    v2f a, b;
    a.x = Al[l * IN_DIM + ka];
    a.y = Al[l * IN_DIM + ka + 1];
    b.x = Wl[ka * HIDDEN + l];
    b.y = Wl[(ka + 1) * HIDDEN + l];
    c = __builtin_amdgcn_wmma_f32_16x16x4_f32(false, a, false, b,
                                              (short)0, c, false, false);
  }

  // D layout: VGPR i -> row (i + 8*half), col l. Apply ReLU and store.
  float* hb = h + (size_t)tile * 16 * HIDDEN;
  #pragma unroll
  for (int i = 0; i < 8; ++i) {
    float v = c[i];
    hb[(i + 8 * half) * HIDDEN + l] = v > 0.0f ? v : 0.0f;
  }
}

// ---------------------------------------------------------------------------
// GEMM2 + log_softmax: out_tile[16x40] = lsm(z2_tile[16x16] @ W2[16x40])
// N padded to 48 (3 WMMA N-tiles), K=16 in 4 steps. Raw scores staged in LDS,
// then lanes 0-15 each reduce one row (max / sum-exp / store).
__global__ void __launch_bounds__(32)
gcn_gemm2_lsm(const float* __restrict__ z2,
              const float* __restrict__ w2,
              float* __restrict__ out) {
  __shared__ float Wl[HIDDEN * NPAD];  // 16x48, zero-padded cols 40..47
  __shared__ float Al[16 * HIDDEN];    // 16x16
  __shared__ float Ol[16 * NPAD];      // raw scores
  const int lane = threadIdx.x;
  const int tile = blockIdx.x;

  for (int i = lane; i < HIDDEN * NPAD; i += 32) {
    int r = i / NPAD, n = i % NPAD;
    Wl[i] = (n < NCLASS) ? w2[r * NCLASS + n] : 0.0f;
  }
  const float* Ab = z2 + (size_t)tile * 16 * HIDDEN;
  for (int i = lane; i < 16 * HIDDEN; i += 32) Al[i] = Ab[i];
  __syncthreads();

  const int half = lane >> 4;
  const int l    = lane & 15;

  #pragma unroll
  for (int t = 0; t < 3; ++t) {
    v8f c = {};
    #pragma unroll
    for (int k0 = 0; k0 < HIDDEN; k0 += 4) {
      const int ka = k0 + 2 * half;
      v2f a, b;
      a.x = Al[l * HIDDEN + ka];
      a.y = Al[l * HIDDEN + ka + 1];
      b.x = Wl[ka * NPAD + t * 16 + l];
      b.y = Wl[(ka + 1) * NPAD + t * 16 + l];
      c = __builtin_amdgcn_wmma_f32_16x16x4_f32(false, a, false, b,
                                                (short)0, c, false, false);
    }
    #pragma unroll
    for (int i = 0; i < 8; ++i)
      Ol[(i + 8 * half) * NPAD + t * 16 + l] = c[i];
  }
  __syncthreads();

  if (lane < 16) {
    const float* row = Ol + lane * NPAD;
    float mx = row[0];
    for (int j = 1; j < NCLASS; ++j) mx = fmaxf(mx, row[j]);
    float s = 0.0f;
    for (int j = 0; j < NCLASS; ++j) s += __expf(row[j] - mx);
    const float lse = mx + __logf(s);
    float* ob = out + ((size_t)tile * 16 + lane) * NCLASS;
    for (int j = 0; j < NCLASS; ++j) ob[j] = row[j] - lse;
  }
}

// ---------------------------------------------------------------------------
extern "C" void kernel_launch(void* const* d_in, const int* in_sizes, int n_in,
                              void* d_out, int out_size, void* d_ws, size_t ws_size,
                              hipStream_t stream) {
  const float* features = (const float*)d_in[0];  // [N, 128]
  const float* weight1  = (const float*)d_in[1];  // [128, 16]
  const float* weight2  = (const float*)d_in[2];  // [16, 40]
  const float* edgeVals = (const float*)d_in[3];  // [E]
  const int*   edgeRows = (const int*)  d_in[4];  // [E]
  const int*   edgeCols = (const int*)  d_in[5];  // [E]

  const int nNodes = in_sizes[0] / IN_DIM;        // 50000
  const int nEdges = in_sizes[3];                 // 800000
  const int nTiles = nNodes / 16;                 // 3125 (exact)

  // Workspace layout (f32): z1 [N*128] | h [N*16] | z2 [N*16]
  float* z1 = (float*)d_ws;
  float* h  = z1 + (size_t)nNodes * IN_DIM;
  float* z2 = h  + (size_t)nNodes * HIDDEN;

  // 1) zero z1..z2 region (h overwritten anyway; zeroing whole span is simplest)
  {
    int n = nNodes * (IN_DIM + 2 * HIDDEN);
    gcn_zero_f32<<<(n + 255) / 256, 256, 0, stream>>>(z1, n);
  }
  // 2) z1 = A @ X
  {
    long long threads = (long long)nEdges * 32;
    gcn_spmm1<<<(unsigned)((threads + 255) / 256), 256, 0, stream>>>(
        edgeRows, edgeCols, edgeVals, features, z1, nEdges);
  }
  // 3) h = relu(z1 @ W1)   [WMMA f32]
  gcn_gemm1_relu<<<nTiles, 32, 0, stream>>>(z1, weight1, h);
  // 4) z2 = A @ h
  {
    long long threads = (long long)nEdges * 4;
    gcn_spmm2<<<(unsigned)((threads + 255) / 256), 256, 0, stream>>>(
        edgeRows, edgeCols, edgeVals, h, z2, nEdges);
  }
  // 5) out = log_softmax(z2 @ W2)   [WMMA f32]
  gcn_gemm2_lsm<<<nTiles, 32, 0, stream>>>(z2, weight2, (float*)d_out);
}